// ChamferDistance_47768626266585
// MI455X (gfx1250) — compile-verified
//
#include <hip/hip_runtime.h>
#include <hip/hip_bf16.h>

typedef __attribute__((ext_vector_type(2))) float v2f;
typedef __attribute__((ext_vector_type(8))) float v8f;

// Each wave owns TILES row-tiles of 16 points from set A; iterates over all of
// set B in 16-point tiles, computing 16x16 squared-distance tiles with
// V_WMMA_F32_16X16X4_F32 (K=4 trick: [-2x0,-2x1,-2x2,1] . [y0,y1,y2,|y|^2]).
#define TILES 2
#define WAVES_PER_BLOCK 8
#define ROWS_PER_BLOCK (16 * TILES * WAVES_PER_BLOCK)  // 256

__global__ __launch_bounds__(256) void
chamfer_dir_kernel(const float* __restrict__ A,   // [B, NA, 3] query points
                   const float* __restrict__ Bp,  // [B, NB, 3] target points
                   float* __restrict__ dist,      // [B, NA]
                   int* __restrict__ idx,         // [B, NA]
                   int NA, int NB) {
  const int lane = threadIdx.x & 31;
  const int wave = threadIdx.x >> 5;
  const int bat  = blockIdx.y;

  const float* __restrict__ Ab = A  + (size_t)bat * NA * 3;
  const float* __restrict__ Bb = Bp + (size_t)bat * NB * 3;

  const int n  = lane & 15;   // column-within-tile / row-within-tile (A side)
  const int hi = lane >> 4;   // which K-half this lane holds (K=0,1 vs K=2,3)

  const int rowBase = (blockIdx.x * WAVES_PER_BLOCK + wave) * 16 * TILES;

  // ---- Build A operands: lane holds row m = tileBase + n, K-pair per half.
  // Row vector = [-2x, -2y, -2z, 1]
  v2f a[TILES];
#pragma unroll
  for (int t = 0; t < TILES; ++t) {
    const int m = rowBase + t * 16 + n;
    const float ax = Ab[m * 3 + 0];
    const float ay = Ab[m * 3 + 1];
    const float az = Ab[m * 3 + 2];
    a[t].x = hi ? (-2.0f * az) : (-2.0f * ax);
    a[t].y = hi ? 1.0f         : (-2.0f * ay);
  }

  float minv[TILES][8];
  int   mini[TILES][8];
#pragma unroll
  for (int t = 0; t < TILES; ++t)
#pragma unroll
    for (int v = 0; v < 8; ++v) { minv[t][v] = __builtin_inff(); mini[t][v] = 0; }

  // ---- Sweep target set in 16-point tiles.
  for (int j0 = 0; j0 < NB; j0 += 16) {
    const int j = j0 + n;
    const float bx = Bb[j * 3 + 0];
    const float by = Bb[j * 3 + 1];
    const float bz = Bb[j * 3 + 2];
    const float sq = bx * bx + by * by + bz * bz;
    // Column vector = [y0, y1, y2, |y|^2], K-pair per lane half.
    v2f bv;
    bv.x = hi ? bz : bx;
    bv.y = hi ? sq : by;

#pragma unroll
    for (int t = 0; t < TILES; ++t) {
      v8f c = {};
      // D[m][n] = -2 * dot(x_m, y_n) + |y_n|^2    (|x_m|^2 added at write-out;
      // it is row-constant so it does not affect the argmin over n)
      v8f d = __builtin_amdgcn_wmma_f32_16x16x4_f32(
          /*neg_a=*/false, a[t], /*neg_b=*/false, bv,
          /*c_mod=*/(short)0, c, /*reuse_a=*/false, /*reuse_b=*/false);
#pragma unroll
      for (int v = 0; v < 8; ++v) {
        const float s = d[v];
        if (s < minv[t][v]) { minv[t][v] = s; mini[t][v] = j; }
      }
    }
  }

  // ---- Reduce (min, argmin) across the 16 lanes of each half-wave.
  // VGPR v in lane (n, hi) corresponds to row rowBase + t*16 + v + 8*hi.
#pragma unroll
  for (int t = 0; t < TILES; ++t) {
#pragma unroll
    for (int v = 0; v < 8; ++v) {
      float mv = minv[t][v];
      int   mi = mini[t][v];
#pragma unroll
      for (int off = 8; off >= 1; off >>= 1) {
        const float ov = __shfl_xor(mv, off, 32);
        const int   oi = __shfl_xor(mi, off, 32);
        if (ov < mv || (ov == mv && oi < mi)) { mv = ov; mi = oi; }
      }
      minv[t][v] = mv;
      mini[t][v] = mi;
    }
  }

  // ---- Lane 0 writes rows +0..7, lane 16 writes rows +8..15 of each tile.
  if (n == 0) {
#pragma unroll
    for (int t = 0; t < TILES; ++t) {
#pragma unroll
      for (int v = 0; v < 8; ++v) {
        const int row = rowBase + t * 16 + v + 8 * hi;
        const float px = Ab[row * 3 + 0];
        const float py = Ab[row * 3 + 1];
        const float pz = Ab[row * 3 + 2];
        const float sqa = px * px + py * py + pz * pz;
        dist[(size_t)bat * NA + row] = minv[t][v] + sqa;
        idx[(size_t)bat * NA + row]  = mini[t][v];
      }
    }
  }
}

extern "C" void kernel_launch(void* const* d_in, const int* in_sizes, int n_in,
                              void* d_out, int out_size, void* d_ws, size_t ws_size,
                              hipStream_t stream) {
  (void)n_in; (void)d_ws; (void)ws_size; (void)out_size;

  const float* xyz1 = (const float*)d_in[0];  // [B, N, 3]
  const float* xyz2 = (const float*)d_in[1];  // [B, M, 3]

  const int BATCH = 4;
  const int BN = in_sizes[0] / 3;  // B*N
  const int BM = in_sizes[1] / 3;  // B*M
  const int N = BN / BATCH;
  const int M = BM / BATCH;

  // d_out layout (reference return order): dist1[B*N], dist2[B*M], idx1[B*N], idx2[B*M]
  float* dist1 = (float*)d_out;
  float* dist2 = dist1 + BN;
  int*   idx1  = (int*)(dist2 + BM);
  int*   idx2  = idx1 + BN;

  dim3 block(32 * WAVES_PER_BLOCK);

  // Direction 1: for each xyz1 point, nearest in xyz2.
  dim3 grid1((N + ROWS_PER_BLOCK - 1) / ROWS_PER_BLOCK, BATCH);
  chamfer_dir_kernel<<<grid1, block, 0, stream>>>(xyz1, xyz2, dist1, idx1, N, M);

  // Direction 2: for each xyz2 point, nearest in xyz1.
  dim3 grid2((M + ROWS_PER_BLOCK - 1) / ROWS_PER_BLOCK, BATCH);
  chamfer_dir_kernel<<<grid2, block, 0, stream>>>(xyz2, xyz1, dist2, idx2, M, N);
}